// RSDPF_62629213110841
// MI455X (gfx1250) — compile-verified
//
#include <hip/hip_runtime.h>
#include <math.h>

// Regime-switching particle filter for MI455X (gfx1250).
// One persistent workgroup per batch (B=32 blocks, 256 thr = 8 wave32 waves,
// 4 waves/SIMD in CU mode). The 63-step scan runs inside the kernel; softmax /
// ESS / cumsum / systematic resample use workgroup barriers only.
// Gatherable per-step arrays (m_t, s_t, cum-weights) live in ~150KB of dynamic
// LDS with diagonal padding so both the interleaved compute pattern and the
// contiguous scan pattern are bank-conflict free. Per-particle carry (m,s,w)
// lives in d_ws, indexed interleaved (q = k*256+tid) so it is thread-private
// AND lane-coalesced.

#define PF_B       32
#define PF_NP      16384
#define PF_T       64
#define PF_NM      8
#define PF_THREADS 256
#define PF_NWAVES  (PF_THREADS / 32)      // 8
#define PF_PPT     (PF_NP / PF_THREADS)   // 64 particles per thread

// diagonal LDS padding: one extra word / 4 extra bytes per 64 particles
__device__ __forceinline__ int FPAD(int p) { return p + (p >> 6); }
__device__ __forceinline__ int BPAD(int p) { return p + ((p >> 6) << 2); }
#define PF_WBUF_WORDS 16640   // FPAD(16383)+1 rounded
#define PF_MA_BYTES   17408   // BPAD(16383)+1 rounded to 16B

__global__ __launch_bounds__(PF_THREADS)
void pf_persist_kernel(const float* __restrict__ o,        // (B,1,T)
                       const int*   __restrict__ init_m,   // (B,NP)
                       const float* __restrict__ init_s,   // (B,NP)
                       const float* __restrict__ u_dyn,    // (B,NP,T)
                       const float* __restrict__ noise,    // (B,NP,T)
                       const float* __restrict__ u_res,    // (B,T)
                       const float* __restrict__ mat_P,    // (NM,NM)
                       const float* __restrict__ co_A,
                       const float* __restrict__ co_B,
                       const float* __restrict__ co_C,
                       const float* __restrict__ co_D,
                       const float* __restrict__ sigma_u,
                       const float* __restrict__ sigma_v,
                       float* __restrict__ out,            // m_list | s_list | w_list
                       int*   __restrict__ m_cur,          // d_ws carry (by q)
                       float* __restrict__ s_cur,
                       float* __restrict__ w_cur)
{
    extern __shared__ unsigned char smem[];
    float*         wbuf = (float*)smem;                          // logw -> e -> cum_e
    float*         sA   = wbuf + PF_WBUF_WORDS;                  // s_t (pre-resample)
    unsigned char* mA   = (unsigned char*)(sA + PF_WBUF_WORDS);  // m_t (pre-resample)

    __shared__ float s_cumP[PF_NM * PF_NM];   // cumulative transition rows
    __shared__ float s_co[4][PF_NM];          // A,B,C,D
    __shared__ float s_red[PF_NWAVES];
    __shared__ float s_red2[PF_NWAVES];
    __shared__ float s_b[4];                  // broadcast: max, S, S2

    const int b    = blockIdx.x;
    const int tid  = threadIdx.x;
    const int lane = tid & 31;
    const int wave = tid >> 5;

    // constants into LDS
    if (tid < PF_NM * PF_NM) {
        const int r = tid >> 3, j = tid & 7;
        float c = 0.f;
        for (int k = 0; k <= j; ++k) c += mat_P[r * PF_NM + k];
        s_cumP[tid] = c;
    }
    if (tid < PF_NM) {
        s_co[0][tid] = co_A[tid];
        s_co[1][tid] = co_B[tid];
        s_co[2][tid] = co_C[tid];
        s_co[3][tid] = co_D[tid];
    }
    const float su     = sigma_u[0];
    const float sv     = sigma_v[0];
    const float inv_sv = 1.0f / sv;
    const float llc  = -logf(sv) - 0.91893853320467274f;  // -log(sv)-0.5*log(2pi)
    const float invN = 1.0f / (float)PF_NP;

    const size_t TOT     = (size_t)PF_B * PF_NP * PF_T;
    const size_t baseIn  = (size_t)b * PF_NP;
    const size_t baseOut = (size_t)b * PF_NP * (size_t)PF_T;

    // ---- t = 0: emit initial state, seed carry (interleaved ownership) ----
    for (int k = 0; k < PF_PPT; ++k) {
        const int q  = k * PF_THREADS + tid;
        const int   m0 = init_m[baseIn + q];
        const float s0 = init_s[baseIn + q];
        m_cur[baseIn + q] = m0;
        s_cur[baseIn + q] = s0;
        w_cur[baseIn + q] = invN;
        const size_t oi = baseOut + (size_t)q * PF_T;
        out[oi]             = (float)m0;
        out[TOT + oi]       = s0;
        out[2 * TOT + oi]   = invN;
    }
    __syncthreads();

    // ---- sequential scan over time ----
    for (int t = 1; t < PF_T; ++t) {
        const float ot = o[b * PF_T + t];
        const float ur = u_res[b * PF_T + t];

        // Phase A: propagate + log-likelihood (interleaved q)
        float lmax = -3.4e38f;
        for (int k = 0; k < PF_PPT; ++k) {
            const int q  = k * PF_THREADS + tid;
            const int   mp = m_cur[baseIn + q];
            const float sp = s_cur[baseIn + q];
            const float wp = w_cur[baseIn + q];
            const size_t gi = (baseIn + (size_t)q) * PF_T + t;
            const float u  = u_dyn[gi];
            const float nz = noise[gi];
            if (((t & 31) == 31) && (t + 1 < PF_T)) {
                // next 128B cacheline of the stride-256B particle streams
                __builtin_prefetch(&u_dyn[gi + 1], 0, 3);
                __builtin_prefetch(&noise[gi + 1], 0, 3);
            }
            // categorical draw: m_t = #(u >= cumP[mp][j]), clipped to NM-1
            const float* cp = &s_cumP[mp * PF_NM];
            int mt = 0;
            #pragma unroll
            for (int j = 0; j < PF_NM; ++j) mt += (u >= cp[j]) ? 1 : 0;
            if (mt > PF_NM - 1) mt = PF_NM - 1;

            const float st   = s_co[0][mt] * sp + s_co[1][mt] + nz * su;
            const float mean = s_co[2][mt] * sqrtf(fabsf(st)) + s_co[3][mt];
            const float d    = (ot - mean) * inv_sv;
            const float lw   = logf(wp + 1e-30f) - 0.5f * d * d + llc;

            mA[BPAD(q)]   = (unsigned char)mt;
            sA[FPAD(q)]   = st;
            wbuf[FPAD(q)] = lw;
            lmax = fmaxf(lmax, lw);
        }

        // max-reduce (wave32 shuffles + wave partials)
        #pragma unroll
        for (int off = 16; off; off >>= 1)
            lmax = fmaxf(lmax, __shfl_xor(lmax, off, 32));
        if (lane == 0) s_red[wave] = lmax;
        __syncthreads();
        if (wave == 0) {
            float v = (lane < PF_NWAVES) ? s_red[lane] : -3.4e38f;
            #pragma unroll
            for (int off = 16; off; off >>= 1)
                v = fmaxf(v, __shfl_xor(v, off, 32));
            if (lane == 0) s_b[0] = v;
        }
        __syncthreads();
        const float Mx = s_b[0];

        // exp + sums (S for softmax, S2 for ESS); order-independent
        float Sl = 0.f, S2l = 0.f;
        for (int k = 0; k < PF_PPT; ++k) {
            const int fq = FPAD(k * PF_THREADS + tid);
            const float e = expf(wbuf[fq] - Mx);
            wbuf[fq] = e;
            Sl  += e;
            S2l += e * e;
        }
        float Sr = Sl, S2r = S2l;
        #pragma unroll
        for (int off = 16; off; off >>= 1) {
            Sr  += __shfl_xor(Sr,  off, 32);
            S2r += __shfl_xor(S2r, off, 32);
        }
        if (lane == 0) { s_red[wave] = Sr; s_red2[wave] = S2r; }
        __syncthreads();
        if (wave == 0) {
            float a = (lane < PF_NWAVES) ? s_red[lane]  : 0.f;
            float c = (lane < PF_NWAVES) ? s_red2[lane] : 0.f;
            #pragma unroll
            for (int off = 16; off; off >>= 1) {
                a += __shfl_xor(a, off, 32);
                c += __shfl_xor(c, off, 32);
            }
            if (lane == 0) { s_b[1] = a; s_b[2] = c; }
        }
        __syncthreads();
        const float S  = s_b[1];
        const float S2 = s_b[2];
        // ess_low:  1/sum(w^2) < N   <=>   S^2 < N * sum(e^2)
        const bool  ess_low = (S * S) < ((float)PF_NP * S2);
        const float invS    = 1.0f / S;

        // weight output + carry (resampling never gathers w)
        for (int k = 0; k < PF_PPT; ++k) {
            const int q = k * PF_THREADS + tid;
            const float wn = ess_low ? invN : wbuf[FPAD(q)] * invS;
            w_cur[baseIn + q] = wn;
            out[2 * TOT + baseOut + (size_t)q * PF_T + t] = wn;
        }

        // inclusive scan of e in particle order -> wbuf becomes cum_e.
        // Scan phase uses CONTIGUOUS chunks (any thread may touch any LDS);
        // diagonal padding keeps this pattern conflict-free too.
        const int p0 = tid * PF_PPT;
        float csum = 0.f;
        for (int k = 0; k < PF_PPT; ++k) csum += wbuf[FPAD(p0 + k)];
        float sc = csum;
        #pragma unroll
        for (int off = 1; off < 32; off <<= 1) {
            const float n = __shfl_up(sc, off, 32);
            if (lane >= off) sc += n;
        }
        if (lane == 31) s_red[wave] = sc;    // wave totals (8 entries)
        __syncthreads();
        float woff = 0.f;
        for (int j = 0; j < wave; ++j) woff += s_red[j];
        float run = woff + sc - csum;        // exclusive prefix for this chunk
        for (int k = 0; k < PF_PPT; ++k) {
            const int fp = FPAD(p0 + k);
            run += wbuf[fp];
            wbuf[fp] = run;
        }
        __syncthreads();

        // resample (systematic) or pass-through (interleaved q)
        if (ess_low) {
            const float scale = S * invN;    // compare in unnormalized cum_e space
            for (int k = 0; k < PF_PPT; ++k) {
                const int q = k * PF_THREADS + tid;
                const float v = ((float)q + ur) * scale;
                // lower_bound: first idx with cum_e[idx] >= v
                int lo = 0, hi = PF_NP;
                while (lo < hi) {
                    const int mid = (lo + hi) >> 1;
                    if (wbuf[FPAD(mid)] < v) lo = mid + 1; else hi = mid;
                }
                const int idx = (lo < PF_NP) ? lo : (PF_NP - 1);
                const int   mn = (int)mA[BPAD(idx)];
                const float sn = sA[FPAD(idx)];
                m_cur[baseIn + q] = mn;
                s_cur[baseIn + q] = sn;
                const size_t oi = baseOut + (size_t)q * PF_T + t;
                out[oi]       = (float)mn;
                out[TOT + oi] = sn;
            }
        } else {
            for (int k = 0; k < PF_PPT; ++k) {
                const int q = k * PF_THREADS + tid;
                const int   mn = (int)mA[BPAD(q)];
                const float sn = sA[FPAD(q)];
                m_cur[baseIn + q] = mn;
                s_cur[baseIn + q] = sn;
                const size_t oi = baseOut + (size_t)q * PF_T + t;
                out[oi]       = (float)mn;
                out[TOT + oi] = sn;
            }
        }
        __syncthreads();   // protect mA/sA/wbuf before next step overwrites
    }
}

extern "C" void kernel_launch(void* const* d_in, const int* in_sizes, int n_in,
                              void* d_out, int out_size, void* d_ws, size_t ws_size,
                              hipStream_t stream) {
    const float* o       = (const float*)d_in[0];
    const int*   init_m  = (const int*)  d_in[1];
    const float* init_s  = (const float*)d_in[2];
    const float* u_dyn   = (const float*)d_in[3];
    const float* noise   = (const float*)d_in[4];
    const float* u_res   = (const float*)d_in[5];
    const float* mat_P   = (const float*)d_in[6];
    const float* co_A    = (const float*)d_in[7];
    const float* co_B    = (const float*)d_in[8];
    const float* co_C    = (const float*)d_in[9];
    const float* co_D    = (const float*)d_in[10];
    const float* sigma_u = (const float*)d_in[11];
    const float* sigma_v = (const float*)d_in[12];

    // carry arrays in workspace: 3 * B*NP * 4B = 6 MB
    char* ws = (char*)d_ws;
    int*   m_cur = (int*)  (ws);
    float* s_cur = (float*)(ws + (size_t)4 * PF_B * PF_NP);
    float* w_cur = (float*)(ws + (size_t)8 * PF_B * PF_NP);

    const size_t shbytes = (size_t)PF_WBUF_WORDS * 4   // wbuf (padded)
                         + (size_t)PF_WBUF_WORDS * 4   // sA   (padded)
                         + (size_t)PF_MA_BYTES;        // mA   (padded) ~150.5 KB

    pf_persist_kernel<<<PF_B, PF_THREADS, shbytes, stream>>>(
        o, init_m, init_s, u_dyn, noise, u_res, mat_P,
        co_A, co_B, co_C, co_D, sigma_u, sigma_v,
        (float*)d_out, m_cur, s_cur, w_cur);
}